// ViGBlock_41489384079613
// MI455X (gfx1250) — compile-verified
//
#include <hip/hip_runtime.h>

typedef __attribute__((ext_vector_type(2))) float v2f;
typedef __attribute__((ext_vector_type(8))) float v8f;

// ---------------------------------------------------------------------------
// LDS-staged fp32 WMMA GEMM:
//   out[nrows,O] = A[nrows,K] @ W[O,K]^T (+ A2 @ W2^T) + bias
// Block = 128 threads = 4 wave32s; block tile = 64 rows x 64 cols.
// K is consumed in 32-wide chunks staged into LDS:
//   - global side: fully coalesced global_load_b128 (8 float4 per thread-chunk)
//   - LDS side:    padded stride 36 floats -> 16B-aligned rows, bank-conflict-
//                  free ds_load_b64 fragment reads (m*36 mod 64 all distinct)
// WMMA op: V_WMMA_F32_16X16X4_F32 (full-precision fp32 matrix path).
// Fragment layouts per CDNA5 ISA:
//   A-frag: lane<16 -> (M=lane, K={k,k+1}); lane>=16 -> (M=lane-16, K={k+2,k+3})
//   B-frag: same with N in place of M
//   C/D:    VGPR r: lanes 0-15 -> (M=r, N=lane); lanes 16-31 -> (M=r+8, N=lane-16)
// ---------------------------------------------------------------------------
#define BK 32
#define LDS_STRIDE 36

__global__ __launch_bounds__(128)
void gemm_wmma_f32(const float* __restrict__ A, const float* __restrict__ W,
                   const float* __restrict__ A2, const float* __restrict__ W2,
                   const float* __restrict__ bias, float* __restrict__ out,
                   int nrows, int K, int O)
{
    __shared__ float As[64 * LDS_STRIDE];
    __shared__ float Bs[64 * LDS_STRIDE];

    const int tid  = threadIdx.x;
    const int lane = tid & 31;
    const int wave = tid >> 5;
    const int row0 = blockIdx.y * 64;
    const int col0 = blockIdx.x * 64;
    const int m    = lane & 15;
    const int koff = (lane >> 4) << 1;     // 0 or 2

    v8f acc0 = {}, acc1 = {}, acc2 = {}, acc3 = {};

    // LDS fragment base pointers (generic ptr to __shared__ -> ds_load)
    const float* as_base  = &As[(wave * 16 + m) * LDS_STRIDE + koff];
    const float* bs_base0 = &Bs[( 0 + m) * LDS_STRIDE + koff];
    const float* bs_base1 = &Bs[(16 + m) * LDS_STRIDE + koff];
    const float* bs_base2 = &Bs[(32 + m) * LDS_STRIDE + koff];
    const float* bs_base3 = &Bs[(48 + m) * LDS_STRIDE + koff];

    const float* Ap = A;
    const float* Wp = W;
    for (int pass = 0; pass < 2; ++pass) {
        if (pass == 1) {
            if (A2 == nullptr) break;       // block-uniform
            Ap = A2; Wp = W2;
        }
        for (int kk = 0; kk < K; kk += BK) {
            __syncthreads();                // protect LDS panels from previous chunk
            // ---- stage 64x32 A-panel and 64x32 B-panel (coalesced b128) ----
            #pragma unroll
            for (int j = 0; j < 4; ++j) {
                const int idx = tid + 128 * j;       // 0..511
                const int r   = idx >> 3;            // 0..63
                const int kq  = (idx & 7) << 2;      // 0,4,...,28
                int gr = row0 + r;                   // clamp ragged rows
                if (gr > nrows - 1) gr = nrows - 1;
                const float4 va = *(const float4*)(Ap + (size_t)gr * K + kk + kq);
                *(float4*)&As[r * LDS_STRIDE + kq] = va;
                const int gc = col0 + r;             // O is a multiple of 64
                const float4 vb = *(const float4*)(Wp + (size_t)gc * K + kk + kq);
                *(float4*)&Bs[r * LDS_STRIDE + kq] = vb;
            }
            __syncthreads();
            // ---- consume chunk: 8 K-steps x 4 col-tiles = 32 WMMAs ----
            #pragma unroll
            for (int k0 = 0; k0 < BK; k0 += 4) {
                v2f a  = *(const v2f*)(as_base  + k0);
                v2f b0 = *(const v2f*)(bs_base0 + k0);
                v2f b1 = *(const v2f*)(bs_base1 + k0);
                v2f b2 = *(const v2f*)(bs_base2 + k0);
                v2f b3 = *(const v2f*)(bs_base3 + k0);
                acc0 = __builtin_amdgcn_wmma_f32_16x16x4_f32(false, a, false, b0, (short)0, acc0, false, false);
                acc1 = __builtin_amdgcn_wmma_f32_16x16x4_f32(false, a, false, b1, (short)0, acc1, false, false);
                acc2 = __builtin_amdgcn_wmma_f32_16x16x4_f32(false, a, false, b2, (short)0, acc2, false, false);
                acc3 = __builtin_amdgcn_wmma_f32_16x16x4_f32(false, a, false, b3, (short)0, acc3, false, false);
            }
        }
    }

    // ---- epilogue: add bias, predicated store (ragged last row-tile) ----
    const int n  = lane & 15;
    const int mb = (lane >> 4) << 3;   // 0 or 8
    const int wrow0 = row0 + wave * 16;
    const float bs0 = bias ? bias[col0 +  0 + n] : 0.f;
    const float bs1 = bias ? bias[col0 + 16 + n] : 0.f;
    const float bs2 = bias ? bias[col0 + 32 + n] : 0.f;
    const float bs3 = bias ? bias[col0 + 48 + n] : 0.f;
    #pragma unroll
    for (int r = 0; r < 8; ++r) {
        const int row = wrow0 + mb + r;
        if (row < nrows) {
            float* p = out + (size_t)row * O + col0;
            p[ 0 + n] = acc0[r] + bs0;
            p[16 + n] = acc1[r] + bs1;
            p[32 + n] = acc2[r] + bs2;
            p[48 + n] = acc3[r] + bs3;
        }
    }
}

// ---------------------------------------------------------------------------
// Edge scatter-add:  agg[dst[e], :] += t[src[e], :]   (256 channels)
// One thread per (edge, 4-channel group): float4 gather + 4 fp32 global atomics.
// Random-destination segment-sum: global_atomic_add_f32 is the right tool;
// this phase is atomic-throughput-bound (~205M atomics).
// ---------------------------------------------------------------------------
__global__ __launch_bounds__(256)
void scatter_add_edges(const float* __restrict__ t, const int* __restrict__ src,
                       const int* __restrict__ dst, float* __restrict__ agg,
                       long long total /* = E * 64 */)
{
    long long idx    = (long long)blockIdx.x * blockDim.x + threadIdx.x;
    long long stride = (long long)gridDim.x * blockDim.x;
    for (; idx < total; idx += stride) {
        const long long e  = idx >> 6;              // / (256/4)
        const int       c4 = (int)(idx & 63) << 2;  // channel group * 4
        const int s = src[e];
        const int d = dst[e];
        const float4 v = *(const float4*)(t + (size_t)s * 256 + c4);
        float* p = agg + (size_t)d * 256 + c4;
        atomicAdd(p + 0, v.x);
        atomicAdd(p + 1, v.y);
        atomicAdd(p + 2, v.z);
        atomicAdd(p + 3, v.w);
    }
}

// ---------------------------------------------------------------------------
// Column sums / sums-of-squares for BatchNorm batch stats.
// Each block reduces a 128-row chunk over all O columns; coalesced loads,
// register accumulation, one atomic per column per block (~200K atomics total).
// ---------------------------------------------------------------------------
__global__ __launch_bounds__(256)
void col_stats(const float* __restrict__ X, int nrows, int O, float* __restrict__ st)
{
    const int r0 = blockIdx.x * 128;
    int r1 = r0 + 128; if (r1 > nrows) r1 = nrows;
    for (int c = threadIdx.x; c < O; c += 256) {
        float s = 0.f, q = 0.f;
        for (int r = r0; r < r1; ++r) {
            const float v = X[(size_t)r * O + c];
            s += v;
            q += v * v;
        }
        atomicAdd(&st[c], s);
        atomicAdd(&st[O + c], q);
    }
}

__global__ void bn_finalize(const float* __restrict__ st, const float* __restrict__ gamma,
                            const float* __restrict__ beta, int O, float invN,
                            float* __restrict__ ss)
{
    const int c = blockIdx.x * blockDim.x + threadIdx.x;
    if (c >= O) return;
    const float mu  = st[c] * invN;
    const float var = st[O + c] * invN - mu * mu;   // biased batch variance
    const float rs  = rsqrtf(var + 1e-5f);
    const float sc  = gamma[c] * rs;
    ss[c]     = sc;
    ss[O + c] = beta[c] - mu * sc;
}

// mode bit0: exact GELU after BN; mode bit1: add residual.
// float4-vectorized: all feature widths (192/256/512) are multiples of 4.
__global__ __launch_bounds__(256)
void bn_apply(const float* __restrict__ X, const float* __restrict__ ss,
              const float* __restrict__ resid, float* __restrict__ Y,
              long long total4 /* = total/4 */, int O, int mode)
{
    long long idx    = (long long)blockIdx.x * blockDim.x + threadIdx.x;
    long long stride = (long long)gridDim.x * blockDim.x;
    for (; idx < total4; idx += stride) {
        const long long base = idx << 2;
        const int c = (int)(base % O);          // base and O both multiples of 4
        const float4 xv = *(const float4*)(X + base);
        const float4 sc = *(const float4*)(ss + c);
        const float4 sh = *(const float4*)(ss + O + c);
        float4 v;
        v.x = xv.x * sc.x + sh.x;
        v.y = xv.y * sc.y + sh.y;
        v.z = xv.z * sc.z + sh.z;
        v.w = xv.w * sc.w + sh.w;
        if (mode & 1) {
            v.x = 0.5f * v.x * (1.f + erff(v.x * 0.70710678118654752f));
            v.y = 0.5f * v.y * (1.f + erff(v.y * 0.70710678118654752f));
            v.z = 0.5f * v.z * (1.f + erff(v.z * 0.70710678118654752f));
            v.w = 0.5f * v.w * (1.f + erff(v.w * 0.70710678118654752f));
        }
        if (mode & 2) {
            const float4 rv = *(const float4*)(resid + base);
            v.x += rv.x; v.y += rv.y; v.z += rv.z; v.w += rv.w;
        }
        *(float4*)(Y + base) = v;
    }
}

// ---------------------------------------------------------------------------
extern "C" void kernel_launch(void* const* d_in, const int* in_sizes, int n_in,
                              void* d_out, int out_size, void* d_ws, size_t ws_size,
                              hipStream_t stream)
{
    constexpr int C = 192, H1 = 256, H2 = 512, HF = 512;

    const float* x        = (const float*)d_in[0];
    const int*   edge     = (const int*)d_in[1];
    const float* w_g1     = (const float*)d_in[2];
    const float* b_g1     = (const float*)d_in[3];
    const float* g1_gamma = (const float*)d_in[4];
    const float* g1_beta  = (const float*)d_in[5];
    const float* w_rel    = (const float*)d_in[6];
    const float* b_rel    = (const float*)d_in[7];
    const float* w_root   = (const float*)d_in[8];
    const float* w_g2     = (const float*)d_in[9];
    const float* b_g2     = (const float*)d_in[10];
    const float* g2_gamma = (const float*)d_in[11];
    const float* g2_beta  = (const float*)d_in[12];
    const float* w_f1     = (const float*)d_in[13];
    const float* b_f1     = (const float*)d_in[14];
    const float* f1_gamma = (const float*)d_in[15];
    const float* f1_beta  = (const float*)d_in[16];
    const float* w_f2     = (const float*)d_in[17];
    const float* b_f2     = (const float*)d_in[18];
    const float* f2_gamma = (const float*)d_in[19];
    const float* f2_beta  = (const float*)d_in[20];

    const int       N = in_sizes[0] / C;
    const long long E = (long long)in_sizes[1] / 2;
    const int* src = edge;          // edge_index[0,:]
    const int* dst = edge + E;      // edge_index[1,:]

    // Workspace layout (floats):
    //   t   : N*H1   (t; later reused for g, then f2)
    //   agg : N*H1   (agg; later reused for x2)
    //   h   : N*H2   (h; later reused for f)
    //   st  : 2*512  (BN sum/sumsq)    ss : 2*512 (BN scale/shift)
    float* ws  = (float*)d_ws;
    float* t   = ws;
    float* agg = t   + (size_t)N * H1;
    float* h   = agg + (size_t)N * H1;
    float* st  = h   + (size_t)N * H2;
    float* ss  = st + 1024;
    float* g   = t;      // reuse after t consumed by GEMM2
    float* x2  = agg;    // reuse after agg consumed by GEMM2
    float* f   = h;      // reuse after h consumed by GEMM3
    float* f2  = t;      // reuse after g consumed by BN-apply

    const int   rowTiles = (N + 63) / 64;
    const float invN     = 1.f / (float)N;
    const dim3  gblk(128);

    auto stats = [&](const float* X, int O, const float* gamma, const float* beta) {
        hipMemsetAsync(st, 0, 2 * (size_t)O * sizeof(float), stream);
        col_stats<<<(N + 127) / 128, 256, 0, stream>>>(X, N, O, st);
        bn_finalize<<<(O + 127) / 128, 128, 0, stream>>>(st, gamma, beta, O, invN, ss);
    };
    auto ewBlocks = [](long long work) {
        return (unsigned)((work + 255) / 256);
    };

    // ---- Grapher ----
    // t = x @ w_g1^T + b_g1                                    [N,256]
    gemm_wmma_f32<<<dim3(H1 / 64, rowTiles), gblk, 0, stream>>>(
        x, w_g1, nullptr, nullptr, b_g1, t, N, C, H1);
    // t = BN(t) (in place)
    stats(t, H1, g1_gamma, g1_beta);
    {
        long long t4 = (long long)N * H1 / 4;
        bn_apply<<<ewBlocks(t4), 256, 0, stream>>>(t, ss, nullptr, t, t4, H1, 0);
    }

    // agg = segment_sum(t[src], dst)                           [N,256]
    hipMemsetAsync(agg, 0, (size_t)N * H1 * sizeof(float), stream);
    {
        long long work = E * (H1 / 4);
        scatter_add_edges<<<ewBlocks(work), 256, 0, stream>>>(t, src, dst, agg, work);
    }

    // h = agg @ w_rel^T + b_rel + t @ w_root^T                 [N,512] (fused dual GEMM)
    gemm_wmma_f32<<<dim3(H2 / 64, rowTiles), gblk, 0, stream>>>(
        agg, w_rel, t, w_root, b_rel, h, N, H1, H2);

    // g = h @ w_g2^T + b_g2                                    [N,192]
    gemm_wmma_f32<<<dim3(C / 64, rowTiles), gblk, 0, stream>>>(
        h, w_g2, nullptr, nullptr, b_g2, g, N, H2, C);
    // x2 = BN(g) + x
    stats(g, C, g2_gamma, g2_beta);
    {
        long long t4 = (long long)N * C / 4;
        bn_apply<<<ewBlocks(t4), 256, 0, stream>>>(g, ss, x, x2, t4, C, 2);
    }

    // ---- FFN ----
    // f = x2 @ w_f1^T + b_f1                                   [N,512]
    gemm_wmma_f32<<<dim3(HF / 64, rowTiles), gblk, 0, stream>>>(
        x2, w_f1, nullptr, nullptr, b_f1, f, N, C, HF);
    // f = gelu(BN(f)) (in place, exact erf GELU)
    stats(f, HF, f1_gamma, f1_beta);
    {
        long long t4 = (long long)N * HF / 4;
        bn_apply<<<ewBlocks(t4), 256, 0, stream>>>(f, ss, nullptr, f, t4, HF, 1);
    }

    // f2 = f @ w_f2^T + b_f2                                   [N,192]
    gemm_wmma_f32<<<dim3(C / 64, rowTiles), gblk, 0, stream>>>(
        f, w_f2, nullptr, nullptr, b_f2, f2, N, HF, C);
    // out = BN(f2) + x2
    stats(f2, C, f2_gamma, f2_beta);
    {
        long long t4 = (long long)N * C / 4;
        bn_apply<<<ewBlocks(t4), 256, 0, stream>>>(f2, ss, x2, (float*)d_out, t4, C, 2);
    }
}